// SerialSpliNet_49933289783798
// MI455X (gfx1250) — compile-verified
//
#include <hip/hip_runtime.h>

// SerialSpliNet: h = (x,x); 20x { h += tanh(W[i]·h + b[i]) }; out = mean(h).
// KS[i] == i for NLAYERS=20, NKNOTS=22 (int(i*21/20) = i for i < 20).
// Compute-bound (tanh-dominated): v_tanh_f32 + v_pk_fma_f32, 8 elems/thread for
// TRANS-latency hiding; branch-free hot kernel + separate tail kernel.

typedef __attribute__((ext_vector_type(2))) float f32x2;
typedef __attribute__((ext_vector_type(4))) float f32x4;

#define NLAYERS 20

__device__ __forceinline__ float fast_tanh(float v) {
#if __has_builtin(__builtin_amdgcn_tanhf)
  return __builtin_amdgcn_tanhf(v);           // gfx1250 v_tanh_f32 (single TRANS op)
#elif __has_builtin(__builtin_amdgcn_tanh_f32)
  return __builtin_amdgcn_tanh_f32(v);
#else
  // tanh(x) = 1 - 2/(exp2(2x*log2e) + 1); saturates correctly at +-inf.
#if __has_builtin(__builtin_amdgcn_exp2f)
  float e = __builtin_amdgcn_exp2f(v * 2.885390081777927f);
#else
  float e = __expf(2.0f * v);
#endif
  return 1.0f - 2.0f * __builtin_amdgcn_rcpf(e + 1.0f);
#endif
}

// One layer: y = {w00,w10}*h.x + {w01,w11}*h.y + {b0,b1}; h += tanh(y) (stepsize==1)
__device__ __forceinline__ f32x2 layer_step(f32x2 h, f32x2 cA, f32x2 cB, f32x2 cb) {
  f32x2 hx = {h.x, h.x};
  f32x2 hy = {h.y, h.y};
  f32x2 y = __builtin_elementwise_fma(cA, hx,
            __builtin_elementwise_fma(cB, hy, cb));  // 2x v_pk_fma_f32
  f32x2 t = {fast_tanh(y.x), fast_tanh(y.y)};        // 2x v_tanh_f32
  return h + t;                                      // v_pk_add_f32
}

// Hot kernel: 8 elements per thread, no divergence, NT b128 streaming.
__global__ __launch_bounds__(256) void splinet_main(
    const float* __restrict__ x, const float* __restrict__ W,
    const float* __restrict__ b, float* __restrict__ out, int noct) {
  int q = blockIdx.x * blockDim.x + threadIdx.x;
  if (q >= noct) return;
  long base = (long)q * 8;

  f32x4 va = __builtin_nontemporal_load((const f32x4*)(x + base));      // global_load_b128
  f32x4 vb = __builtin_nontemporal_load((const f32x4*)(x + base + 4));

  f32x2 h[8];
  h[0] = (f32x2){va.x, va.x};
  h[1] = (f32x2){va.y, va.y};
  h[2] = (f32x2){va.z, va.z};
  h[3] = (f32x2){va.w, va.w};
  h[4] = (f32x2){vb.x, vb.x};
  h[5] = (f32x2){vb.y, vb.y};
  h[6] = (f32x2){vb.z, vb.z};
  h[7] = (f32x2){vb.w, vb.w};

  // W[i] row-major [2][2]: y_r = h0*W[r][0] + h1*W[r][1] + b_r
  // cA = column 0 = {W[4i+0], W[4i+2]}, cB = column 1 = {W[4i+1], W[4i+3]}
#pragma unroll
  for (int i = 0; i < NLAYERS; ++i) {
    f32x2 cA = {W[i * 4 + 0], W[i * 4 + 2]};   // uniform -> s_load, K$ cached
    f32x2 cB = {W[i * 4 + 1], W[i * 4 + 3]};
    f32x2 cb = {b[i * 2 + 0], b[i * 2 + 1]};
#pragma unroll
    for (int e = 0; e < 8; ++e) h[e] = layer_step(h[e], cA, cB, cb);
  }

  f32x4 oa = {0.5f * (h[0].x + h[0].y), 0.5f * (h[1].x + h[1].y),
              0.5f * (h[2].x + h[2].y), 0.5f * (h[3].x + h[3].y)};
  f32x4 ob = {0.5f * (h[4].x + h[4].y), 0.5f * (h[5].x + h[5].y),
              0.5f * (h[6].x + h[6].y), 0.5f * (h[7].x + h[7].y)};
  __builtin_nontemporal_store(oa, (f32x4*)(out + base));                // global_store_b128
  __builtin_nontemporal_store(ob, (f32x4*)(out + base + 4));
}

// Tail kernel: only launched when n % 8 != 0 (never for B = 8388608).
__global__ __launch_bounds__(32) void splinet_tail(
    const float* __restrict__ x, const float* __restrict__ W,
    const float* __restrict__ b, float* __restrict__ out, int start, int n) {
  int i = start + (int)threadIdx.x;
  if (i >= n) return;
  float v = x[i];
  f32x2 h = {v, v};
#pragma unroll
  for (int l = 0; l < NLAYERS; ++l) {
    f32x2 cA = {W[l * 4 + 0], W[l * 4 + 2]};
    f32x2 cB = {W[l * 4 + 1], W[l * 4 + 3]};
    f32x2 cb = {b[l * 2 + 0], b[l * 2 + 1]};
    h = layer_step(h, cA, cB, cb);
  }
  out[i] = 0.5f * (h.x + h.y);
}

extern "C" void kernel_launch(void* const* d_in, const int* in_sizes, int n_in,
                              void* d_out, int out_size, void* d_ws, size_t ws_size,
                              hipStream_t stream) {
  const float* x = (const float*)d_in[0];   // [B,1] f32
  const float* W = (const float*)d_in[1];   // [24,2,2] f32 (only first 20 used)
  const float* b = (const float*)d_in[2];   // [24,2] f32
  float* out = (float*)d_out;               // [B,1] f32

  const int n = in_sizes[0];                // B = 8388608
  const int noct = n / 8;                   // full 8-element groups
  if (noct > 0) {
    const int block = 256;                  // 8 wave32s per workgroup
    const int grid = (noct + block - 1) / block;
    splinet_main<<<grid, block, 0, stream>>>(x, W, b, out, noct);
  }
  const int rem = n - noct * 8;
  if (rem > 0) {
    splinet_tail<<<1, 32, 0, stream>>>(x, W, b, out, noct * 8, n);
  }
}